// ConditionalRandomField_81733227643482
// MI455X (gfx1250) — compile-verified
//
#include <hip/hip_runtime.h>
#include <hip/hip_bf16.h>

#define NT   25
#define NTP  28          // K padded to 7 chunks of 4
#define SEQL 512
#define BATCHN 2048

typedef float v2f __attribute__((ext_vector_type(2)));
typedef float v8f __attribute__((ext_vector_type(8)));

// ---------------------------------------------------------------------------
// Forward (denominator) kernel: one wave (32 lanes) per 16 sequences.
// alpha recurrence via V_WMMA_F32_16X16X4_F32:
//   D[j][b] = sum_i E[i][j] * p[b][i],  E = exp(transitions), A = E^T (const),
//   B[k][n] = p-row of sequence n (lane-local), 7 K-chunks, 2 M-tiles (25 tags).
// ---------------------------------------------------------------------------
__global__ __launch_bounds__(32)
void crf_forward_kernel(const float* __restrict__ logits,
                        const float* __restrict__ trans,
                        const float* __restrict__ startT,
                        const float* __restrict__ endT,
                        const int*  __restrict__ mask,
                        float* __restrict__ ws_den)
{
    __shared__ float emitbuf[16 * NT];

    const int lane = threadIdx.x;          // 0..31
    const int hi   = lane >> 4;            // which half of the wave
    const int n    = lane & 15;            // sequence-in-tile (N column)
    const long tileBase = (long)blockIdx.x * 16;
    const long b = tileBase + n;

    // ---- constant A = E^T in WMMA A-layout (16x4 f32 per chunk) -----------
    // lane (hi,n): A[m=n(+16*tile)][k0=4c+2*hi .. +1]  value = exp(T[k][j])
    v2f a0[7], a1[7];
#pragma unroll
    for (int c = 0; c < 7; ++c) {
        const int k0 = 4 * c + 2 * hi;
        const int j0 = n, j1 = n + 16;
        float x0 = 0.f, y0 = 0.f, x1 = 0.f, y1 = 0.f;
        if (k0 < NT) {
            x0 = __expf(trans[k0 * NT + j0]);
            if (j1 < NT) x1 = __expf(trans[k0 * NT + j1]);
        }
        if (k0 + 1 < NT) {
            y0 = __expf(trans[(k0 + 1) * NT + j0]);
            if (j1 < NT) y1 = __expf(trans[(k0 + 1) * NT + j1]);
        }
        a0[c].x = x0; a0[c].y = y0;
        a1[c].x = x1; a1[c].y = y1;
    }

    // ---- init: alpha = start + emit[t=0] (stage emissions through LDS) ----
    for (int f = lane; f < 16 * NT; f += 32) {
        const int bb = f / NT, j = f % NT;
        emitbuf[f] = logits[((tileBase + bb) * SEQL + 0) * NT + j];
    }
    asm volatile("s_wait_dscnt 0" ::: "memory");

    float alpha[NT];
#pragma unroll
    for (int j = 0; j < NT; ++j)
        alpha[j] = startT[j] + emitbuf[n * NT + j];

    // ---- 511 recurrence steps --------------------------------------------
    for (int t = 1; t < SEQL; ++t) {
        // stage emit[:,t,:] for this 16-sequence tile (coalesced-ish)
        for (int f = lane; f < 16 * NT; f += 32) {
            const int bb = f / NT, j = f % NT;
            emitbuf[f] = logits[((tileBase + bb) * SEQL + t) * NT + j];
        }

        // lane-local row max + p = exp(alpha - m)  (each lane owns row n)
        float m = alpha[0];
#pragma unroll
        for (int j = 1; j < NT; ++j) m = fmaxf(m, alpha[j]);
        float p[NTP];
#pragma unroll
        for (int j = 0; j < NT; ++j) p[j] = __expf(alpha[j] - m);
        p[25] = 0.f; p[26] = 0.f; p[27] = 0.f;

        // 7 K-chunks x 2 M-tiles of V_WMMA_F32_16X16X4_F32
        v8f d0 = {0.f,0.f,0.f,0.f,0.f,0.f,0.f,0.f};
        v8f d1 = {0.f,0.f,0.f,0.f,0.f,0.f,0.f,0.f};
#pragma unroll
        for (int c = 0; c < 7; ++c) {
            v2f bv;
            bv.x = hi ? p[4 * c + 2] : p[4 * c + 0];
            bv.y = hi ? p[4 * c + 3] : p[4 * c + 1];
            d0 = __builtin_amdgcn_wmma_f32_16x16x4_f32(false, a0[c], false, bv,
                                                       (short)0, d0, false, false);
            d1 = __builtin_amdgcn_wmma_f32_16x16x4_f32(false, a1[c], false, bv,
                                                       (short)0, d1, false, false);
        }

        // D layout: lane holds (seq n, tags v+8*hi) / (16+v+8*hi).
        // Exchange lane halves so every lane has the full 25-tag row.
        float x0[8], x1[8];
#pragma unroll
        for (int v = 0; v < 8; ++v) {
            x0[v] = __shfl_xor((float)d0[v], 16, 32);
            x1[v] = __shfl_xor((float)d1[v], 16, 32);
        }
        float s[NT];
#pragma unroll
        for (int v = 0; v < 8; ++v) {
            s[v]      = hi ? x0[v]        : (float)d0[v];
            s[8 + v]  = hi ? (float)d0[v] : x0[v];
            s[16 + v] = hi ? x1[v]        : (float)d1[v];
        }
        s[24] = hi ? (float)d1[0] : x1[0];

        asm volatile("s_wait_dscnt 0" ::: "memory");
        const float mm = (float)mask[b * SEQL + t];
        const float qq = 1.0f - mm;
#pragma unroll
        for (int j = 0; j < NT; ++j) {
            const float upd = m + __logf(s[j]) + emitbuf[n * NT + j];
            alpha[j] = upd * mm + alpha[j] * qq;
        }
    }

    // ---- stops: logsumexp(alpha + end) -----------------------------------
    float st[NT];
#pragma unroll
    for (int j = 0; j < NT; ++j) st[j] = alpha[j] + endT[j];
    float m2 = st[0];
#pragma unroll
    for (int j = 1; j < NT; ++j) m2 = fmaxf(m2, st[j]);
    float sum = 0.f;
#pragma unroll
    for (int j = 0; j < NT; ++j) sum += __expf(st[j] - m2);
    if (hi == 0) ws_den[b] = m2 + __logf(sum);
}

// ---------------------------------------------------------------------------
// Numerator (joint likelihood): one block per sequence, deterministic tree
// reduction of emission/transition scores + mask count.
// ---------------------------------------------------------------------------
__global__ __launch_bounds__(256)
void crf_joint_kernel(const float* __restrict__ logits,
                      const float* __restrict__ trans,
                      const float* __restrict__ startT,
                      const float* __restrict__ endT,
                      const long long* __restrict__ tags,
                      const int* __restrict__ mask,
                      float* __restrict__ ws_num)
{
    __shared__ float redf[256];
    __shared__ int   redi[256];
    const int tid = threadIdx.x;
    const long b = blockIdx.x;
    const long long* tagr = tags + b * SEQL;
    const int*   mr = mask + b * SEQL;
    const float* lr = logits + b * SEQL * NT;

    float acc = 0.f;
    int cnt = 0;
    for (int t = tid; t < SEQL; t += 256) {
        const int mt = mr[t];
        cnt += mt;
        const float mf = (float)mt;
        const int tg = (int)tagr[t];
        if (t >= 1) {
            const int pv = (int)tagr[t - 1];
            acc += trans[pv * NT + tg] * mf;           // trans_score, mask[:,1:]
        }
        if (t <= SEQL - 2)
            acc += lr[t * NT + tg] * mf;               // emit_score, mask[:,:-1]
    }
    redf[tid] = acc; redi[tid] = cnt;
    __syncthreads();
    for (int s = 128; s > 0; s >>= 1) {
        if (tid < s) { redf[tid] += redf[tid + s]; redi[tid] += redi[tid + s]; }
        __syncthreads();
    }
    if (tid == 0) {
        float score = redf[0];
        score += startT[(int)tagr[0]];
        const int li = redi[0] - 1;                    // last valid index
        const int lt = (int)tagr[li];
        const float mfl = (float)mr[SEQL - 1];
        score += endT[lt] + lr[(SEQL - 1) * NT + lt] * mfl;
        ws_num[b] = score;
    }
}

// ---------------------------------------------------------------------------
// Final deterministic reduction: sum_b (num[b] - den[b]) -> d_out[0]
// ---------------------------------------------------------------------------
__global__ __launch_bounds__(256)
void crf_reduce_kernel(const float* __restrict__ ws_den,
                       const float* __restrict__ ws_num,
                       float* __restrict__ out)
{
    __shared__ float red[256];
    const int tid = threadIdx.x;
    float acc = 0.f;
    for (int b = tid; b < BATCHN; b += 256)
        acc += ws_num[b] - ws_den[b];
    red[tid] = acc;
    __syncthreads();
    for (int s = 128; s > 0; s >>= 1) {
        if (tid < s) red[tid] += red[tid + s];
        __syncthreads();
    }
    if (tid == 0) out[0] = red[0];
}

extern "C" void kernel_launch(void* const* d_in, const int* in_sizes, int n_in,
                              void* d_out, int out_size, void* d_ws, size_t ws_size,
                              hipStream_t stream)
{
    const float*     logits = (const float*)d_in[0];      // (2048,512,25) f32
    const float*     trans  = (const float*)d_in[1];      // (25,25) f32
    const float*     startT = (const float*)d_in[2];      // (25,) f32
    const float*     endT   = (const float*)d_in[3];      // (25,) f32
    const long long* tags   = (const long long*)d_in[4];  // (2048,512) i64
    const int*       maskp  = (const int*)d_in[5];        // (2048,512) i32
    float* out = (float*)d_out;

    float* ws_den = (float*)d_ws;          // [0 .. 2047]
    float* ws_num = ws_den + BATCHN;       // [2048 .. 4095]

    crf_forward_kernel<<<BATCHN / 16, 32, 0, stream>>>(logits, trans, startT, endT,
                                                       maskp, ws_den);
    crf_joint_kernel<<<BATCHN, 256, 0, stream>>>(logits, trans, startT, endT,
                                                 tags, maskp, ws_num);
    crf_reduce_kernel<<<1, 256, 0, stream>>>(ws_den, ws_num, out);
}